// ParaLIF_14439680049302
// MI455X (gfx1250) — compile-verified
//
#include <hip/hip_runtime.h>
#include <hip/hip_bf16.h>

typedef __attribute__((ext_vector_type(2))) float v2f;
typedef __attribute__((ext_vector_type(8))) float v8f;

#define BB   16
#define TT   2048
#define II   256
#define HH   512
#define MTOT (BB * TT)   // 32768 rows

// alpha = beta = exp(-0.1) in fp32
#define BETA_F   0.90483743f
#define OMBETA_F (1.0f - 0.90483743f)

// ---------------------------------------------------------------------------
// GEMM1: X[m,h] = sum_k x[m,k] * W[h,k] + b[h]
// One wave -> 32(M) x 64(N): 2 A fragments x 4 B fragments -> 8 WMMA / k-step,
// 6 b64 loads / k-step (0.75 loads per v_wmma_f32_16x16x4_f32).
// Block = 8 waves = full 512-wide N for one 32-row M tile (shared A rows).
// ---------------------------------------------------------------------------
__global__ __launch_bounds__(256) void gemm1_wmma(
    const float* __restrict__ x, const float* __restrict__ W,
    const float* __restrict__ bias, float* __restrict__ X)
{
    const int mt    = blockIdx.x;           // 0..1023 (32 rows each)
    const int ng    = threadIdx.x >> 5;     // wave id 0..7 -> 64-col group
    const int lane  = threadIdx.x & 31;
    const int lr    = lane & 15;
    const int kHalf = (lane >> 4) << 1;     // lanes 0-15: K=0,1  lanes 16-31: K=2,3
    const int nbase = ng * 64;

    const float* __restrict__ A0 = x + (size_t)(mt * 32 + lr) * II;
    const float* __restrict__ A1 = x + (size_t)(mt * 32 + 16 + lr) * II;
    const float* __restrict__ B0 = W + (size_t)(nbase + 0 * 16 + lr) * II;
    const float* __restrict__ B1 = W + (size_t)(nbase + 1 * 16 + lr) * II;
    const float* __restrict__ B2 = W + (size_t)(nbase + 2 * 16 + lr) * II;
    const float* __restrict__ B3 = W + (size_t)(nbase + 3 * 16 + lr) * II;

    v8f acc[8] = {};   // acc[2*j + i]: n-tile j, m-half i
    for (int k = 0; k < II; k += 4) {
        v2f a0 = *(const v2f*)(A0 + k + kHalf);
        v2f a1 = *(const v2f*)(A1 + k + kHalf);
        v2f b0 = *(const v2f*)(B0 + k + kHalf);
        v2f b1 = *(const v2f*)(B1 + k + kHalf);
        v2f b2 = *(const v2f*)(B2 + k + kHalf);
        v2f b3 = *(const v2f*)(B3 + k + kHalf);
        acc[0] = __builtin_amdgcn_wmma_f32_16x16x4_f32(false, a0, false, b0, (short)0, acc[0], false, false);
        acc[1] = __builtin_amdgcn_wmma_f32_16x16x4_f32(false, a1, false, b0, (short)0, acc[1], false, false);
        acc[2] = __builtin_amdgcn_wmma_f32_16x16x4_f32(false, a0, false, b1, (short)0, acc[2], false, false);
        acc[3] = __builtin_amdgcn_wmma_f32_16x16x4_f32(false, a1, false, b1, (short)0, acc[3], false, false);
        acc[4] = __builtin_amdgcn_wmma_f32_16x16x4_f32(false, a0, false, b2, (short)0, acc[4], false, false);
        acc[5] = __builtin_amdgcn_wmma_f32_16x16x4_f32(false, a1, false, b2, (short)0, acc[5], false, false);
        acc[6] = __builtin_amdgcn_wmma_f32_16x16x4_f32(false, a0, false, b3, (short)0, acc[6], false, false);
        acc[7] = __builtin_amdgcn_wmma_f32_16x16x4_f32(false, a1, false, b3, (short)0, acc[7], false, false);
    }

    // D layout: VGPR r -> M = r + 8*(lane>=16), N = lane&15
    const int rbase = (lane >> 4) << 3;
    #pragma unroll
    for (int j = 0; j < 4; ++j) {
        const int h = nbase + j * 16 + lr;
        const float bv = bias[h];
        #pragma unroll
        for (int i = 0; i < 2; ++i) {
            float* __restrict__ outp = X + (size_t)(mt * 32 + i * 16 + rbase) * HH + h;
            const v8f a = acc[2 * j + i];
            #pragma unroll
            for (int r = 0; r < 8; ++r)
                outp[(size_t)r * HH] = a[r] + bv;
        }
    }
}

// ---------------------------------------------------------------------------
// Scan 1: cascaded first-order IIRs == the reference FFT leaky integration
// (k[t]=beta^t(1-beta), l[t]=alpha^t; rfft(n=2T)->irfft[:T] is causal conv).
// One thread per (b,h); consecutive lanes -> consecutive h (coalesced).
// ---------------------------------------------------------------------------
__global__ __launch_bounds__(256) void scan1_iir(
    const float* __restrict__ X, float* __restrict__ mem1)
{
    const int tid = blockIdx.x * blockDim.x + threadIdx.x;   // 0..8191
    const int b = tid >> 9;
    const int h = tid & (HH - 1);
    size_t idx = (size_t)b * TT * HH + h;
    float syn = 0.0f, mem = 0.0f;
    #pragma unroll 4
    for (int t = 0; t < TT; ++t, idx += HH) {
        syn = BETA_F * syn + OMBETA_F * X[idx];
        mem = BETA_F * mem + syn;
        mem1[idx] = mem;
    }
}

// ---------------------------------------------------------------------------
// GEMM2: recur_in[m,g] = X[m,g] + br[g] + sum_h relu(mem1[m-1,h]) * Wr[g,h]
// One-step shift handled on the A-load path with a 0/1 mask multiply
// (addresses clamped valid; no branch -> EXEC stays all-ones for WMMA).
// ---------------------------------------------------------------------------
__global__ __launch_bounds__(256) void gemm2_wmma(
    const float* __restrict__ mem1, const float* __restrict__ Wr,
    const float* __restrict__ br, const float* __restrict__ X,
    float* __restrict__ rin)
{
    const int mt    = blockIdx.x;           // 0..1023 (32 rows each)
    const int ng    = threadIdx.x >> 5;
    const int lane  = threadIdx.x & 31;
    const int lr    = lane & 15;
    const int kHalf = (lane >> 4) << 1;
    const int nbase = ng * 64;

    const int m0 = mt * 32 + lr;
    const int m1 = m0 + 16;
    const int t0 = m0 & (TT - 1);
    const int t1 = m1 & (TT - 1);
    const float am0 = (t0 == 0) ? 0.0f : 1.0f;     // zero row at t==0
    const float am1 = (t1 == 0) ? 0.0f : 1.0f;
    const int mr0 = (t0 == 0) ? m0 : (m0 - 1);     // clamped valid address
    const int mr1 = (t1 == 0) ? m1 : (m1 - 1);

    const float* __restrict__ A0 = mem1 + (size_t)mr0 * HH;
    const float* __restrict__ A1 = mem1 + (size_t)mr1 * HH;
    const float* __restrict__ B0 = Wr + (size_t)(nbase + 0 * 16 + lr) * HH;
    const float* __restrict__ B1 = Wr + (size_t)(nbase + 1 * 16 + lr) * HH;
    const float* __restrict__ B2 = Wr + (size_t)(nbase + 2 * 16 + lr) * HH;
    const float* __restrict__ B3 = Wr + (size_t)(nbase + 3 * 16 + lr) * HH;

    v8f acc[8] = {};
    for (int k = 0; k < HH; k += 4) {
        v2f a0 = *(const v2f*)(A0 + k + kHalf);
        v2f a1 = *(const v2f*)(A1 + k + kHalf);
        a0.x = fmaxf(a0.x, 0.0f) * am0;            // relu + shift mask
        a0.y = fmaxf(a0.y, 0.0f) * am0;
        a1.x = fmaxf(a1.x, 0.0f) * am1;
        a1.y = fmaxf(a1.y, 0.0f) * am1;
        v2f b0 = *(const v2f*)(B0 + k + kHalf);
        v2f b1 = *(const v2f*)(B1 + k + kHalf);
        v2f b2 = *(const v2f*)(B2 + k + kHalf);
        v2f b3 = *(const v2f*)(B3 + k + kHalf);
        acc[0] = __builtin_amdgcn_wmma_f32_16x16x4_f32(false, a0, false, b0, (short)0, acc[0], false, false);
        acc[1] = __builtin_amdgcn_wmma_f32_16x16x4_f32(false, a1, false, b0, (short)0, acc[1], false, false);
        acc[2] = __builtin_amdgcn_wmma_f32_16x16x4_f32(false, a0, false, b1, (short)0, acc[2], false, false);
        acc[3] = __builtin_amdgcn_wmma_f32_16x16x4_f32(false, a1, false, b1, (short)0, acc[3], false, false);
        acc[4] = __builtin_amdgcn_wmma_f32_16x16x4_f32(false, a0, false, b2, (short)0, acc[4], false, false);
        acc[5] = __builtin_amdgcn_wmma_f32_16x16x4_f32(false, a1, false, b2, (short)0, acc[5], false, false);
        acc[6] = __builtin_amdgcn_wmma_f32_16x16x4_f32(false, a0, false, b3, (short)0, acc[6], false, false);
        acc[7] = __builtin_amdgcn_wmma_f32_16x16x4_f32(false, a1, false, b3, (short)0, acc[7], false, false);
    }

    const int rbase = (lane >> 4) << 3;
    #pragma unroll
    for (int j = 0; j < 4; ++j) {
        const int g = nbase + j * 16 + lr;
        const float bv = br[g];
        #pragma unroll
        for (int i = 0; i < 2; ++i) {
            const size_t base = (size_t)(mt * 32 + i * 16 + rbase) * HH + g;
            const v8f a = acc[2 * j + i];
            #pragma unroll
            for (int r = 0; r < 8; ++r)
                rin[base + (size_t)r * HH] = a[r] + X[base + (size_t)r * HH] + bv;
        }
    }
}

// ---------------------------------------------------------------------------
// Scan 2 + spike: mem2 = mem1 + IIR(recur_in); out = (Δmem2 - thr) > 0.
// rin_out is read then overwritten in place (same thread, same index).
// ---------------------------------------------------------------------------
__global__ __launch_bounds__(256) void scan2_spike(
    const float* __restrict__ mem1, float* __restrict__ rin_out,
    const float* __restrict__ thr)
{
    const int tid = blockIdx.x * blockDim.x + threadIdx.x;
    const int b = tid >> 9;
    const int h = tid & (HH - 1);
    const float th = thr[h];
    size_t idx = (size_t)b * TT * HH + h;
    float syn = 0.0f, mem = 0.0f, prev = 0.0f;
    #pragma unroll 4
    for (int t = 0; t < TT; ++t, idx += HH) {
        syn = BETA_F * syn + OMBETA_F * rin_out[idx];
        mem = BETA_F * mem + syn;
        const float m2 = mem1[idx] + mem;
        rin_out[idx] = ((m2 - prev - th) > 0.0f) ? 1.0f : 0.0f;
        prev = m2;
    }
}

extern "C" void kernel_launch(void* const* d_in, const int* in_sizes, int n_in,
                              void* d_out, int out_size, void* d_ws, size_t ws_size,
                              hipStream_t stream) {
    (void)in_sizes; (void)n_in; (void)out_size;
    const float* x   = (const float*)d_in[0];
    const float* W   = (const float*)d_in[1];
    const float* b   = (const float*)d_in[2];
    const float* Wr  = (const float*)d_in[3];
    const float* br  = (const float*)d_in[4];
    const float* thr = (const float*)d_in[5];
    float* out = (float*)d_out;

    const size_t elems = (size_t)MTOT * HH;          // 16,777,216
    if (ws_size < 2 * elems * sizeof(float)) return; // need X + mem1 in scratch
    float* Xbuf = (float*)d_ws;
    float* mem1 = Xbuf + elems;

    gemm1_wmma <<<MTOT / 32, 256, 0, stream>>>(x, W, b, Xbuf);
    scan1_iir  <<<(BB * HH) / 256, 256, 0, stream>>>(Xbuf, mem1);
    gemm2_wmma <<<MTOT / 32, 256, 0, stream>>>(mem1, Wr, br, Xbuf, out); // out = recur_in (scratch)
    scan2_spike<<<(BB * HH) / 256, 256, 0, stream>>>(mem1, out, thr);
}